// KGAggregator_27642409517753
// MI455X (gfx1250) — compile-verified
//
#include <hip/hip_runtime.h>
#include <hip/hip_bf16.h>
#include <stdint.h>

// ---------------------------------------------------------------------------
// KG scatter-mean aggregation, MI455X (gfx1250, wave32).
//
// Roofline: ~154 MFLOP vs ~0.9 GB of moved data -> bandwidth/atomic bound.
// Naive scatter needs 76.8M global_atomic_add_f32 (~200us at L2 atomic rate),
// 5-10x above the ~25us HBM/L2 floor. So: counting-sort edges by head (1.2M
// int atomics total), then one wave32 per head accumulates its segment in
// registers and writes each output row exactly once (non-temporal). The 100KB
// relation table is staged into LDS with the CDNA5 async global->LDS DMA path
// (ASYNCcnt) and read back via ds_load_b128.
// ---------------------------------------------------------------------------

typedef __attribute__((ext_vector_type(4))) float v4f;

// Stage `n16` float4s into dynamic LDS via GLOBAL_LOAD_ASYNC_TO_LDS_B128
// (GV mode: vdst = per-lane LDS byte offset, vaddr = per-lane 64-bit global
// address). Generic __shared__ pointers are {aperture_hi32, lds_offset32}, so
// the u32 truncation is exactly the VDST LDS byte address.
__device__ __forceinline__ void async_stage_to_lds(float4* __restrict__ sdst,
                                                   const float4* __restrict__ src,
                                                   int n16) {
    const int tid  = threadIdx.x;
    const int nthr = blockDim.x;
    for (int i = tid; i < n16; i += nthr) {   // <=25 outstanding, ASYNCcnt max 63
        unsigned lds_off = (unsigned)(uintptr_t)(&sdst[i]);
        const float4* g = src + i;
        asm volatile("global_load_async_to_lds_b128 %0, %1, off"
                     :: "v"(lds_off), "v"(g)
                     : "memory");
    }
    asm volatile("s_wait_asynccnt 0x0" ::: "memory");
    __syncthreads();
}

// ---------------- Pass 1: per-head edge counts (int atomics) ----------------
__global__ void kg_count_kernel(const int* __restrict__ head,
                                int* __restrict__ counts, int n_edges) {
    const int i = blockIdx.x * blockDim.x + threadIdx.x;
    if (i < n_edges) atomicAdd(&counts[head[i]], 1);
}

// ---------------- Pass 2: exclusive scan -> offsets[N+1], cursor[N] ---------
// Single 1024-thread workgroup; ~N/1024 chunks of LDS Hillis-Steele scan.
__global__ void kg_scan_kernel(const int* __restrict__ counts,
                               int* __restrict__ offsets,
                               int* __restrict__ cursor, int n) {
    __shared__ int s[1024];
    const int tid = threadIdx.x;
    int carry = 0;                                    // uniform across block
    for (int base = 0; base < n; base += 1024) {
        const int i = base + tid;
        const int v = (i < n) ? counts[i] : 0;
        s[tid] = v;
        __syncthreads();
        for (int off = 1; off < 1024; off <<= 1) {
            const int t = (tid >= off) ? s[tid - off] : 0;
            __syncthreads();
            s[tid] += t;
            __syncthreads();
        }
        const int incl = s[tid];
        if (i < n) {
            const int o = carry + incl - v;           // exclusive
            offsets[i] = o;
            cursor[i]  = o;
        }
        const int total = s[1023];
        __syncthreads();                              // protect s[] reuse
        carry += total;
    }
    if (tid == 0) offsets[n] = carry;                 // == n_edges
}

// ---------------- Pass 3: scatter edge ids into segment order ---------------
__global__ void kg_scatter_ids_kernel(const int* __restrict__ head,
                                      int* __restrict__ cursor,
                                      int* __restrict__ sorted, int n_edges) {
    const int i = blockIdx.x * blockDim.x + threadIdx.x;
    if (i < n_edges) {
        const int p = atomicAdd(&cursor[head[i]], 1);
        sorted[p] = i;
    }
}

// ---------------- Pass 4: one wave32 per head, register accumulation --------
__global__ void kg_aggregate_kernel(const float* __restrict__ entity_emb,
                                    const int*   __restrict__ tail,
                                    const int*   __restrict__ etype,
                                    const float* __restrict__ relation_emb,
                                    const int*   __restrict__ offsets,
                                    const int*   __restrict__ sorted,
                                    float* __restrict__ out,
                                    int n_entities, int n_rel) {
    extern __shared__ float4 s_rel[];                 // n_rel rows * 32 float4
    async_stage_to_lds(s_rel, (const float4*)relation_emb, n_rel * 32);

    const int lane   = threadIdx.x & 31;
    const int gwave  = (blockIdx.x * blockDim.x + threadIdx.x) >> 5;
    const int nwaves = (gridDim.x * blockDim.x) >> 5;

    for (int h = gwave; h < n_entities; h += nwaves) {
        const int start = offsets[h];
        const int end   = offsets[h + 1];
        float ax = 0.f, ay = 0.f, az = 0.f, aw = 0.f;

        for (int base = start; base < end; base += 32) {
            const int cnt = min(32, end - base);
            // Coalesced per-lane index prefetch for up to 32 edges.
            const int e  = (lane < cnt) ? sorted[base + lane] : 0;
            const int tl = (lane < cnt) ? tail[e]  : 0;
            const int rl = (lane < cnt) ? etype[e] : 0;
            for (int j = 0; j < cnt; ++j) {
                const int t = __shfl(tl, j, 32);      // ds_bpermute broadcast
                const int r = __shfl(rl, j, 32);
                const float4 tv = ((const float4*)(entity_emb + (size_t)t * 128))[lane];
                const float4 rv = s_rel[r * 32 + lane];   // ds_load_b128
                ax += tv.x * rv.x;
                ay += tv.y * rv.y;
                az += tv.z * rv.z;
                aw += tv.w * rv.w;
            }
        }
        // Fold in the mean; each row written exactly once -> non-temporal
        // store keeps the L2-resident entity table from being evicted.
        const float c = (float)(end - start);
        const float s = 1.0f / fmaxf(c, 1.0f);
        v4f o; o.x = ax * s; o.y = ay * s; o.z = az * s; o.w = aw * s;
        __builtin_nontemporal_store(o, (v4f*)(out + (size_t)h * 128) + lane);
    }
}

// ---------------- Fallback (small ws): direct f32-atomic scatter ------------
__global__ void kg_scatter_atomic_kernel(const float* __restrict__ entity_emb,
                                         const int*   __restrict__ head,
                                         const int*   __restrict__ tail,
                                         const int*   __restrict__ etype,
                                         const float* __restrict__ relation_emb,
                                         float* __restrict__ out_sum,
                                         float* __restrict__ counts,
                                         int n_edges, int n_rel) {
    extern __shared__ float4 s_rel[];
    async_stage_to_lds(s_rel, (const float4*)relation_emb, n_rel * 32);

    const int lane   = threadIdx.x & 31;
    const int gwave  = (blockIdx.x * blockDim.x + threadIdx.x) >> 5;
    const int nwaves = (gridDim.x * blockDim.x) >> 5;

    for (int e = gwave; e < n_edges; e += nwaves) {
        const int h = head[e];
        const int t = tail[e];
        const int r = etype[e];
        const float4 tv = ((const float4*)(entity_emb + (size_t)t * 128))[lane];
        const float4 rv = s_rel[r * 32 + lane];
        float* dst = out_sum + (size_t)h * 128 + lane * 4;
        unsafeAtomicAdd(dst + 0, tv.x * rv.x);
        unsafeAtomicAdd(dst + 1, tv.y * rv.y);
        unsafeAtomicAdd(dst + 2, tv.z * rv.z);
        unsafeAtomicAdd(dst + 3, tv.w * rv.w);
        if (lane == 0) unsafeAtomicAdd(&counts[h], 1.0f);
    }
}

__global__ void kg_normalize_kernel(float* __restrict__ out,
                                    const float* __restrict__ counts,
                                    int n_entities) {
    const int tid   = blockIdx.x * blockDim.x + threadIdx.x;
    const int total = n_entities * 32;
    if (tid >= total) return;
    const int row  = tid >> 5;
    const int lane = tid & 31;
    const float s  = 1.0f / fmaxf(counts[row], 1.0f);
    float4* p = (float4*)(out + (size_t)row * 128) + lane;
    float4 v = *p;
    v.x *= s; v.y *= s; v.z *= s; v.w *= s;
    *p = v;
}

// ---------------------------------------------------------------------------
extern "C" void kernel_launch(void* const* d_in, const int* in_sizes, int n_in,
                              void* d_out, int out_size, void* d_ws, size_t ws_size,
                              hipStream_t stream) {
    // setup_inputs() order:
    //   0: entity_emb [N,128] f32 | 1: edge_index [2,E] int | 2: edge_type [E]
    //   3: relation_emb [R,128] f32 | 4: n_entities scalar
    const float* entity_emb   = (const float*)d_in[0];
    const int*   edge_index   = (const int*)  d_in[1];
    const int*   edge_type    = (const int*)  d_in[2];
    const float* relation_emb = (const float*)d_in[3];

    const int N = in_sizes[0] / 128;
    const int E = in_sizes[2];
    const int R = in_sizes[3] / 128;

    const int* head = edge_index;       // row 0
    const int* tail = edge_index + E;   // row 1

    const size_t lds_bytes = (size_t)R * 128 * sizeof(float);
    float* out = (float*)d_out;

    // ws layout (fast path): counts[N] | offsets[N+1] | cursor[N] | sorted[E]
    const size_t need = ((size_t)3 * N + 1 + (size_t)E) * sizeof(int);

    if (ws_size >= need) {
        int* counts  = (int*)d_ws;
        int* offsets = counts + N;
        int* cursor  = offsets + (N + 1);
        int* sorted  = cursor + N;

        hipMemsetAsync(counts, 0, (size_t)N * sizeof(int), stream);

        const int eb = (E + 255) / 256;
        kg_count_kernel<<<eb, 256, 0, stream>>>(head, counts, E);
        kg_scan_kernel<<<1, 1024, 0, stream>>>(counts, offsets, cursor, N);
        kg_scatter_ids_kernel<<<eb, 256, 0, stream>>>(head, cursor, sorted, E);
        // 2048 blocks x 8 waves, grid-stride over heads: amortizes the 100KB
        // LDS staging to ~200MB of L2 reads while filling every WGP.
        kg_aggregate_kernel<<<2048, 256, lds_bytes, stream>>>(
            entity_emb, tail, edge_type, relation_emb, offsets, sorted,
            out, N, R);
    } else {
        // Fallback: direct atomic scatter (needs only N floats of scratch).
        float* fcounts = (float*)d_ws;
        hipMemsetAsync(d_out, 0, (size_t)N * 128 * sizeof(float), stream);
        hipMemsetAsync(fcounts, 0, (size_t)N * sizeof(float), stream);
        kg_scatter_atomic_kernel<<<1024, 256, lds_bytes, stream>>>(
            entity_emb, head, tail, edge_type, relation_emb, out, fcounts, E, R);
        const int total = N * 32;
        kg_normalize_kernel<<<(total + 255) / 256, 256, 0, stream>>>(out, fcounts, N);
    }
}